// GNNEncoder_19602230739553
// MI455X (gfx1250) — compile-verified
//
#include <hip/hip_runtime.h>

typedef __attribute__((ext_vector_type(2))) float v2f;
typedef __attribute__((ext_vector_type(8))) float v8f;

#define IN_DIM 20
#define HID 64

// ---------------- zeroing ----------------
__global__ void k_zero4(float4* p, unsigned int n4) {
    unsigned int i = blockIdx.x * blockDim.x + threadIdx.x;
    if (i < n4) { float4 z; z.x = z.y = z.z = z.w = 0.0f; p[i] = z; }
}
__global__ void k_zero(float* p, int n) {
    int i = blockIdx.x * blockDim.x + threadIdx.x;
    if (i < n) p[i] = 0.0f;
}

// ------------- layer-1 scatter: agg1[dst] += x[src], deg[dst] += 1 -------------
// One thread per (edge, feature); total = E*20 < 2^31 so 32-bit indexing, and
// /20 by constant lowers to a mulhi reciprocal (no 64-bit divide sequence).
__global__ void k_scatter1(const float* __restrict__ x, const int* __restrict__ src,
                           const int* __restrict__ dst, float* __restrict__ agg1,
                           float* __restrict__ deg, unsigned int total) {
    unsigned int idx = blockIdx.x * blockDim.x + threadIdx.x;
    if (idx >= total) return;
    unsigned int e = idx / IN_DIM;
    int d = (int)(idx - e * IN_DIM);
    int s = src[e];
    int t = dst[e];
    atomicAdd(&agg1[t * IN_DIM + d], x[s * IN_DIM + d]);
    if (d == 0) atomicAdd(&deg[t], 1.0f);
}

// ------------- deg -> 1/deg in place -------------
__global__ void k_deginv(float* deg, int N) {
    int i = blockIdx.x * blockDim.x + threadIdx.x;
    if (i < N) {
        float d = deg[i];
        deg[i] = (d > 0.0f) ? (1.0f / d) : 0.0f;
    }
}

// ------------- layer 1: h1 = relu((agg1*deginv)@W1l + b1 + x@W1r)  (K=20) -------------
// One wave = one 16x16 output tile. 128 threads = 4 waves cover 16 rows x 64 cols.
__global__ void k_layer1(const float* __restrict__ agg1, const float* __restrict__ x,
                         const float* __restrict__ deginv,
                         const float* __restrict__ Wl, const float* __restrict__ bias,
                         const float* __restrict__ Wr,
                         float* __restrict__ h1, int N) {
    const int lane = threadIdx.x & 31;
    const int wv   = threadIdx.x >> 5;      // column tile 0..3
    const int half = lane >> 4;             // 0/1 (K-half for A/B fragments)
    const int m    = lane & 15;
    const int row0 = blockIdx.x * 16;
    int arow = row0 + m;
    if (arow >= N) arow = N - 1;            // clamp (keep EXEC all-ones for WMMA)
    const int col = (wv << 4) + m;

    const float dinv = deginv[arow];
    const float* aptr = agg1 + arow * IN_DIM;
    const float* xptr = x    + arow * IN_DIM;

    v8f acc = {};
#pragma unroll
    for (int ks = 0; ks < IN_DIM / 4; ++ks) {           // agg-path: (agg1*dinv) @ Wl
        const int k = 4 * ks + 2 * half;
        v2f a; a.x = aptr[k] * dinv; a.y = aptr[k + 1] * dinv;
        v2f b; b.x = Wl[k * HID + col]; b.y = Wl[(k + 1) * HID + col];
        acc = __builtin_amdgcn_wmma_f32_16x16x4_f32(false, a, false, b,
                                                    (short)0, acc, false, false);
    }
#pragma unroll
    for (int ks = 0; ks < IN_DIM / 4; ++ks) {           // root-path: x @ Wr
        const int k = 4 * ks + 2 * half;
        v2f a; a.x = xptr[k]; a.y = xptr[k + 1];
        v2f b; b.x = Wr[k * HID + col]; b.y = Wr[(k + 1) * HID + col];
        acc = __builtin_amdgcn_wmma_f32_16x16x4_f32(false, a, false, b,
                                                    (short)0, acc, false, false);
    }
    const float bv = bias[col];
    float* hcol = h1 + (row0 + 8 * half) * HID + col;   // D row = row0 + i + 8*half
    if (row0 + 16 <= N) {                               // uniform: full tile, no guards
#pragma unroll
        for (int i = 0; i < 8; ++i) {
            float v = acc[i] + bv;
            hcol[i * HID] = (v > 0.0f) ? v : 0.0f;
        }
    } else {                                            // tail tile (unused when N%16==0)
#pragma unroll
        for (int i = 0; i < 8; ++i) {
            if (row0 + 8 * half + i < N) {
                float v = acc[i] + bv;
                hcol[i * HID] = (v > 0.0f) ? v : 0.0f;
            }
        }
    }
}

// ------------- layer-2 scatter: agg2[dst] += h1[src] (64 dims) -------------
__global__ void k_scatter2(const float* __restrict__ h1, const int* __restrict__ src,
                           const int* __restrict__ dst, float* __restrict__ agg2,
                           unsigned int total) {
    unsigned int idx = blockIdx.x * blockDim.x + threadIdx.x;
    if (idx >= total) return;
    unsigned int e = idx >> 6;
    int d = (int)(idx & 63u);
    int s = src[e];
    int t = dst[e];
    atomicAdd(&agg2[t * HID + d], h1[s * HID + d]);
}

// ------------- layer 2 + fused column-mean numerator (K=64) -------------
__global__ void k_layer2(const float* __restrict__ agg2, const float* __restrict__ h1,
                         const float* __restrict__ deginv,
                         const float* __restrict__ Wl, const float* __restrict__ bias,
                         const float* __restrict__ Wr,
                         float* __restrict__ out, int N) {
    const int lane = threadIdx.x & 31;
    const int wv   = threadIdx.x >> 5;
    const int half = lane >> 4;
    const int m    = lane & 15;
    const int row0 = blockIdx.x * 16;
    int arow = row0 + m;
    if (arow >= N) arow = N - 1;
    const int col = (wv << 4) + m;

    const float dinv = deginv[arow];
    const float* aptr = agg2 + arow * HID;
    const float* hptr = h1   + arow * HID;

    v8f acc = {};
#pragma unroll
    for (int ks = 0; ks < HID / 4; ++ks) {              // (agg2*dinv) @ W2l
        const int k = 4 * ks + 2 * half;
        v2f a; a.x = aptr[k] * dinv; a.y = aptr[k + 1] * dinv;
        v2f b; b.x = Wl[k * HID + col]; b.y = Wl[(k + 1) * HID + col];
        acc = __builtin_amdgcn_wmma_f32_16x16x4_f32(false, a, false, b,
                                                    (short)0, acc, false, false);
    }
#pragma unroll
    for (int ks = 0; ks < HID / 4; ++ks) {              // h1 @ W2r
        const int k = 4 * ks + 2 * half;
        v2f a; a.x = hptr[k]; a.y = hptr[k + 1];
        v2f b; b.x = Wr[k * HID + col]; b.y = Wr[(k + 1) * HID + col];
        acc = __builtin_amdgcn_wmma_f32_16x16x4_f32(false, a, false, b,
                                                    (short)0, acc, false, false);
    }
    // Sum this lane's 8 rows for its column, bias added per valid row.
    const float bv = bias[col];
    float s = 0.0f;
    if (row0 + 16 <= N) {                               // uniform fast path
#pragma unroll
        for (int i = 0; i < 8; ++i) s += acc[i] + bv;
    } else {
#pragma unroll
        for (int i = 0; i < 8; ++i)
            if (row0 + 8 * half + i < N) s += acc[i] + bv;
    }
    // Combine the two halves holding the same column (wave32 xor-shuffle).
    s += __shfl_xor(s, 16);
    if (half == 0) atomicAdd(&out[col], s);
}

__global__ void k_finalize(float* out, float invN) {
    int i = threadIdx.x;
    if (i < HID) out[i] *= invN;
}

extern "C" void kernel_launch(void* const* d_in, const int* in_sizes, int n_in,
                              void* d_out, int out_size, void* d_ws, size_t ws_size,
                              hipStream_t stream) {
    const float* x   = (const float*)d_in[0];
    const int*   ei  = (const int*)d_in[1];
    const float* W1l = (const float*)d_in[2];
    const float* b1  = (const float*)d_in[3];
    const float* W1r = (const float*)d_in[4];
    const float* W2l = (const float*)d_in[5];
    const float* b2  = (const float*)d_in[6];
    const float* W2r = (const float*)d_in[7];

    const int N = in_sizes[0] / IN_DIM;
    const int E = in_sizes[1] / 2;
    const int* src = ei;
    const int* dst = ei + E;

    // workspace layout: deg[N] | agg1[N*20] | agg2[N*64] | h1[N*64]
    float* ws   = (float*)d_ws;
    float* deg  = ws;
    float* agg1 = deg + N;
    float* agg2 = agg1 + (size_t)N * IN_DIM;
    float* h1   = agg2 + (size_t)N * HID;
    float* out  = (float*)d_out;

    // zero deg + agg1 + agg2 (h1 is fully overwritten; float4 overshoot lands in h1)
    const unsigned int zf = (unsigned int)N * (1 + IN_DIM + HID);
    const unsigned int z4 = (zf + 3) / 4;
    k_zero4<<<(z4 + 255) / 256, 256, 0, stream>>>((float4*)ws, z4);
    k_zero<<<1, 64, 0, stream>>>(out, HID);

    const unsigned int t1 = (unsigned int)E * IN_DIM;       // 64M < 2^31
    k_scatter1<<<(t1 + 255) / 256, 256, 0, stream>>>(x, src, dst, agg1, deg, t1);
    k_deginv<<<(N + 255) / 256, 256, 0, stream>>>(deg, N);

    const int nTiles = (N + 15) / 16;
    k_layer1<<<nTiles, 128, 0, stream>>>(agg1, x, deg, W1l, b1, W1r, h1, N);

    const unsigned int t2 = (unsigned int)E * HID;          // 204.8M < 2^31
    k_scatter2<<<(t2 + 255) / 256, 256, 0, stream>>>(h1, src, dst, agg2, t2);

    k_layer2<<<nTiles, 128, 0, stream>>>(agg2, h1, deg, W2l, b2, W2r, out, N);
    k_finalize<<<1, 64, 0, stream>>>(out, 1.0f / (float)N);
}